// GCN_9070970929449
// MI455X (gfx1250) — compile-verified
//
#include <hip/hip_runtime.h>

typedef __attribute__((ext_vector_type(2))) float v2f;
typedef __attribute__((ext_vector_type(8))) float v8f;

static constexpr int NN     = 100000;   // nodes
static constexpr int NE     = 1600000;  // edges
static constexpr int DK     = 128;      // inner dim of every GEMM (D_IN == D_HID == 128)
static constexpr int MTILES = NN / 16;  // 6250 exact

// ---------------------------------------------------------------------------
// Dense GEMM:  out[N, DOUT] = (relu?)A[N,128] @ W[128,DOUT] + bias
// One wave32 computes a 16-row slab x all DOUT columns using
// V_WMMA_F32_16X16X4_F32 (f32 A/B, f32 accum), K stepped by 4.
//
// f32 16x16x4 VGPR layouts (ISA 7.12.2):
//   A (16x4):  lane = M (lanes 0-15 and 16-31 both M=0..15),
//              {v0,v1} = K = {0,1} for lanes 0-15, {2,3} for lanes 16-31
//   B (4x16):  lane = N, same K split across lane halves (mirror of A)
//   C/D (16x16): lane = N, VGPR v holds M = v + 8*(lane>=16)
// ---------------------------------------------------------------------------
template<int NT, bool RELU_IN>
__global__ __launch_bounds__(128)
void gemm_wmma_f32(const float* __restrict__ A, const float* __restrict__ W,
                   const float* __restrict__ bias, float* __restrict__ out)
{
    constexpr int DOUT = NT * 16;
    const int lane  = threadIdx.x & 31;
    const int wave  = threadIdx.x >> 5;
    const int mtile = blockIdx.x * 4 + wave;
    if (mtile >= MTILES) return;          // uniform per wave; tiles divide exactly

    const int half = lane >> 4;           // 0: lanes 0-15, 1: lanes 16-31
    const int mn   = lane & 15;           // M index (A) / N index (B,C,D)
    const int kb   = half * 2;            // K sub-offset inside each 4-wide step

    const float* arow = A + (size_t)(mtile * 16 + mn) * DK;

    v8f acc[NT];
#pragma unroll
    for (int t = 0; t < NT; ++t) acc[t] = (v8f)0.0f;

    for (int k = 0; k < DK; k += 4) {
        v2f a = *(const v2f*)(arow + k + kb);          // 8B aligned (k%4==0, kb even)
        if (RELU_IN) { a.x = fmaxf(a.x, 0.0f); a.y = fmaxf(a.y, 0.0f); }
        const float* wp = W + (size_t)(k + kb) * DOUT + mn;
#pragma unroll
        for (int t = 0; t < NT; ++t) {
            v2f b;
            b.x = wp[t * 16];
            b.y = wp[t * 16 + DOUT];
            acc[t] = __builtin_amdgcn_wmma_f32_16x16x4_f32(
                false, a, false, b, (short)0, acc[t], false, false);
        }
    }

#pragma unroll
    for (int t = 0; t < NT; ++t) {
        const float bv = bias[t * 16 + mn];
#pragma unroll
        for (int v = 0; v < 8; ++v) {
            const int m = mtile * 16 + v + 8 * half;
            out[(size_t)m * DOUT + t * 16 + mn] = acc[t][v] + bv;
        }
    }
}

// ---------------------------------------------------------------------------
// SpMM: out[row[e]] += vals[e] * x[col[e]]   (edge-parallel, one wave per edge)
// D = 32*VEC floats per row; lane covers VEC consecutive floats.
// Coalesced 512B/256B row gather (L2-resident: 51.2MB << 192MB L2),
// non-returning global_atomic_add_f32 for the scatter.
// ---------------------------------------------------------------------------
template<int VEC>
__global__ __launch_bounds__(256)
void spmm_atomic(const float* __restrict__ vals, const int* __restrict__ row,
                 const int* __restrict__ col, const float* __restrict__ x,
                 float* __restrict__ out)
{
    constexpr int D = 32 * VEC;
    const long long gt = (long long)blockIdx.x * blockDim.x + threadIdx.x;
    const int e    = (int)(gt >> 5);
    const int lane = (int)(gt & 31);
    if (e >= NE) return;                  // uniform per wave (NE % 8 == 0 anyway)

    const float v = vals[e];
    const int   r = row[e];
    const int   c = col[e];

    const float* xs = x   + (size_t)c * D + lane * VEC;
    float*       od = out + (size_t)r * D + lane * VEC;

    float t[VEC];
    if constexpr (VEC == 4) {
        const float4 q = *(const float4*)xs;
        t[0] = q.x; t[1] = q.y; t[2] = q.z; t[3] = q.w;
    } else {
        const float2 q = *(const float2*)xs;
        t[0] = q.x; t[1] = q.y;
    }
#pragma unroll
    for (int i = 0; i < VEC; ++i)
        atomicAdd(od + i, v * t[i]);
}

// ---------------------------------------------------------------------------
extern "C" void kernel_launch(void* const* d_in, const int* in_sizes, int n_in,
                              void* d_out, int out_size, void* d_ws, size_t ws_size,
                              hipStream_t stream)
{
    const float* x    = (const float*)d_in[0];
    const float* vals = (const float*)d_in[1];
    const float* W1   = (const float*)d_in[2];
    const float* b1   = (const float*)d_in[3];
    const float* W2   = (const float*)d_in[4];
    const float* b2   = (const float*)d_in[5];
    const float* W3   = (const float*)d_in[6];
    const float* b3   = (const float*)d_in[7];
    const int*   row  = (const int*)d_in[8];
    const int*   col  = (const int*)d_in[9];

    float* bufA = (float*)d_ws;                 // [NN,128] 51.2 MB
    float* bufB = bufA + (size_t)NN * 128;      // [NN,128] 51.2 MB

    const dim3 gBlk(128);                       // 4 waves / block
    const dim3 gGrd((MTILES + 3) / 4);          // 1563 blocks
    const dim3 sBlk(256);                       // 8 edges / block
    const dim3 sGrd(NE / 8);                    // 200000 blocks

    // Layer 1: bufA = x@W1+b1 ; bufB = spmm(bufA)
    gemm_wmma_f32<8, false><<<gGrd, gBlk, 0, stream>>>(x, W1, b1, bufA);
    hipMemsetAsync(bufB, 0, (size_t)NN * 128 * sizeof(float), stream);
    spmm_atomic<4><<<sGrd, sBlk, 0, stream>>>(vals, row, col, bufA, bufB);

    // Layer 2: bufA = relu(bufB)@W2+b2 ; bufB = spmm(bufA)
    gemm_wmma_f32<8, true><<<gGrd, gBlk, 0, stream>>>(bufB, W2, b2, bufA);
    hipMemsetAsync(bufB, 0, (size_t)NN * 128 * sizeof(float), stream);
    spmm_atomic<4><<<sGrd, sBlk, 0, stream>>>(vals, row, col, bufA, bufB);

    // Layer 3: bufA[:, :64] = relu(bufB)@W3+b3 ; out = spmm(bufA) (no relu)
    gemm_wmma_f32<4, true><<<gGrd, gBlk, 0, stream>>>(bufB, W3, b3, bufA);
    hipMemsetAsync(d_out, 0, (size_t)NN * 64 * sizeof(float), stream);
    spmm_atomic<2><<<sGrd, sBlk, 0, stream>>>(vals, row, col, bufA, (float*)d_out);
}